// SubgraphPooling_82995948028007
// MI455X (gfx1250) — compile-verified
//
#include <hip/hip_runtime.h>

typedef unsigned int u32;
typedef float f32x4 __attribute__((ext_vector_type(4)));  // native vector: OK for nontemporal builtins

#define DFEAT  128
#define SCAN_B 1024

// ---------------- Phase 1: zero segment counts ----------------
__global__ void k_zero_counts(u32* __restrict__ counts, int nseg) {
    int i = blockIdx.x * blockDim.x + threadIdx.x;
    if (i < nseg) counts[i] = 0u;
}

// ---------------- Phase 2: histogram of segment ids ----------------
// seg[] is a once-read 8MB stream -> NT load keeps L2 for the hot regions.
__global__ void k_hist(const int* __restrict__ seg, u32* __restrict__ counts, int nedge) {
    int e = blockIdx.x * blockDim.x + threadIdx.x;
    if (e < nedge) {
        int s = __builtin_nontemporal_load(seg + e);
        atomicAdd(&counts[s], 1u);
    }
}

// ---------------- Phase 3a: per-block exclusive scan ----------------
__global__ __launch_bounds__(SCAN_B)
void k_scan1(const u32* __restrict__ counts, u32* __restrict__ offsets,
             u32* __restrict__ bsums, int n) {
    __shared__ u32 sh[SCAN_B];
    const int t   = threadIdx.x;
    const int idx = blockIdx.x * SCAN_B + t;
    u32 v = (idx < n) ? counts[idx] : 0u;
    sh[t] = v;
    __syncthreads();
    for (int off = 1; off < SCAN_B; off <<= 1) {
        u32 x = (t >= off) ? sh[t - off] : 0u;
        __syncthreads();
        sh[t] += x;
        __syncthreads();
    }
    if (idx < n) offsets[idx] = sh[t] - v;           // exclusive prefix in block
    if (t == SCAN_B - 1) bsums[blockIdx.x] = sh[t];  // block total
}

// ---------------- Phase 3b: scan the block sums (nblk <= 1024) ----------------
__global__ __launch_bounds__(SCAN_B)
void k_scan2(u32* __restrict__ bsums, int nblk) {
    __shared__ u32 sh[SCAN_B];
    const int t = threadIdx.x;
    u32 v = (t < nblk) ? bsums[t] : 0u;
    sh[t] = v;
    __syncthreads();
    for (int off = 1; off < SCAN_B; off <<= 1) {
        u32 x = (t >= off) ? sh[t - off] : 0u;
        __syncthreads();
        sh[t] += x;
        __syncthreads();
    }
    if (t < nblk) bsums[t] = sh[t] - v;  // exclusive
}

// ---------------- Phase 3c: add block bases; init scatter cursors ----------------
__global__ __launch_bounds__(SCAN_B)
void k_scan3(u32* __restrict__ offsets, const u32* __restrict__ bsums,
             u32* __restrict__ cursor, int n) {
    const int idx = blockIdx.x * SCAN_B + threadIdx.x;
    if (idx < n) {
        u32 o = offsets[idx] + bsums[blockIdx.x];
        offsets[idx] = o;
        cursor[idx]  = o;
    }
}

// ---------------- Phase 4: bucket the NODE ids by segment (CSR payload) ----------------
__global__ void k_scatter(const int* __restrict__ seg, const int* __restrict__ nodeids,
                          u32* __restrict__ cursor, u32* __restrict__ perm, int nedge) {
    int e = blockIdx.x * blockDim.x + threadIdx.x;
    if (e < nedge) {
        int s   = __builtin_nontemporal_load(seg + e);
        int nid = __builtin_nontemporal_load(nodeids + e);
        u32 pos = atomicAdd(&cursor[s], 1u);
        perm[pos] = (u32)nid;   // scattered write; stays in L2, re-read once in phase 5
    }
}

// ---------------- Phase 5: one wave32 per segment: sum rows, divide, stream out --------
// Lane l owns 4 consecutive floats (f32x4) of the 128-wide row: 32 lanes x 16B = 512B,
// i.e. one fully coalesced global_load_b128 per lane per row, served from the 192MB L2
// (feature table is only 51MB and stays resident; output is written nontemporally so the
// 128MB result stream does not evict it).
__global__ void k_segmean(const float* __restrict__ feat,
                          const u32* __restrict__ offsets,
                          const u32* __restrict__ counts,
                          const u32* __restrict__ perm,
                          float* __restrict__ out, int nseg) {
    const int lane = threadIdx.x & 31;
    const int wave = threadIdx.x >> 5;
    const int s    = blockIdx.x * (blockDim.x >> 5) + wave;
    if (s >= nseg) return;

    const u32 base = offsets[s];
    const int cnt  = (int)counts[s];

    f32x4 acc = {0.f, 0.f, 0.f, 0.f};

    for (int c0 = 0; c0 < cnt; c0 += 32) {
        const int rem = cnt - c0;
        const int m   = rem < 32 ? rem : 32;
        u32 id = 0u;
        if (lane < m) id = perm[base + c0 + lane];  // coalesced chunk of node ids

        // software pipeline: one ds_bpermute (wave32 broadcast) per row; the freshly
        // broadcast id is prefetched (global_prefetch_b8) one iteration ahead so the
        // next 512B row is already in flight from L2 while this one accumulates.
        u32 nid = (u32)__builtin_amdgcn_ds_bpermute(0, (int)id);
        __builtin_prefetch(feat + (size_t)nid * DFEAT + lane * 4, 0, 3);
        for (int i = 0; i < m; ++i) {
            u32 nid_next = nid;
            if (i + 1 < m) {
                nid_next = (u32)__builtin_amdgcn_ds_bpermute((i + 1) << 2, (int)id);
                __builtin_prefetch(feat + (size_t)nid_next * DFEAT + lane * 4, 0, 3);
            }
            const f32x4 v = *reinterpret_cast<const f32x4*>(
                feat + (size_t)nid * DFEAT + lane * 4);
            acc += v;
            nid = nid_next;
        }
    }

    const float inv = 1.0f / (float)(cnt > 1 ? cnt : 1);  // empty segment -> 0/1 = 0
    f32x4 r = acc * inv;
    // nontemporal: 128MB result stream must not evict the hot feature table from L2
    __builtin_nontemporal_store(r, reinterpret_cast<f32x4*>(
        out + (size_t)s * DFEAT + lane * 4));
}

extern "C" void kernel_launch(void* const* d_in, const int* in_sizes, int n_in,
                              void* d_out, int out_size, void* d_ws, size_t ws_size,
                              hipStream_t stream) {
    const float* feat    = (const float*)d_in[0];  // [N_NODES, 128] f32
    const int*   nodeids = (const int*)d_in[1];    // [N_GATHER] int
    const int*   seg     = (const int*)d_in[2];    // [N_GATHER] int
    float*       out     = (float*)d_out;          // [N_SEGMENTS, 128] f32

    const int nedge = in_sizes[1];
    const int nseg  = out_size / DFEAT;

    // workspace layout (u32 units): counts | offsets | cursor | bsums(1024) | perm
    u32* counts  = (u32*)d_ws;
    u32* offsets = counts + nseg;
    u32* cursor  = offsets + nseg;
    u32* bsums   = cursor + nseg;
    u32* perm    = bsums + 1024;

    const int nblk_scan = (nseg + SCAN_B - 1) / SCAN_B;  // 245 for nseg=250000 (<=1024)

    k_zero_counts<<<(nseg + 255) / 256, 256, 0, stream>>>(counts, nseg);
    k_hist<<<(nedge + 255) / 256, 256, 0, stream>>>(seg, counts, nedge);
    k_scan1<<<nblk_scan, SCAN_B, 0, stream>>>(counts, offsets, bsums, nseg);
    k_scan2<<<1, SCAN_B, 0, stream>>>(bsums, nblk_scan);
    k_scan3<<<nblk_scan, SCAN_B, 0, stream>>>(offsets, bsums, cursor, nseg);
    k_scatter<<<(nedge + 255) / 256, 256, 0, stream>>>(seg, nodeids, cursor, perm, nedge);

    const int waves_per_block = 8;  // 256 threads = 8 wave32
    k_segmean<<<(nseg + waves_per_block - 1) / waves_per_block, waves_per_block * 32, 0,
                stream>>>(feat, offsets, counts, perm, out, nseg);
}